// TrainableODE_22170621182732
// MI455X (gfx1250) — compile-verified
//
#include <hip/hip_runtime.h>
#include <stdint.h>

// Smith CVS model, Heun integrator.
// B=2048 systems, one per lane; 64 single-wave (wave32) workgroups.
// Snapshots go LDS -> global via CDNA5 async stores (ASYNCcnt), double-buffered.

#define ODE_DT       0.002f
#define ODE_SUBSTEPS 20

struct SmithP {
    float E_lv, E_rv, E_ao, E_vc, E_pa, E_pu;
    float rR_mt, rR_av, rR_sys, rR_tc, rR_pv, rR_pul;   // reciprocals (hoisted divides)
    float P0_l, lam_l, P0_r, lam_r;
};

// periodic elastance driver e(t); wave-uniform but computed per lane (VALU+TRANS)
__device__ __forceinline__ float smith_driver(float t) {
    float tm = t - 0.75f * floorf(t * (1.0f / 0.75f));
    float x  = tm - 0.27f;
    return __expf(-80.0f * x * x);
}

// vector field given precomputed driver value e
__device__ __forceinline__ void smith_vf(float e, const float v[6], float d[6],
                                         const SmithP& p) {
    float oe   = 1.0f - e;
    float P_lv = e * p.E_lv * v[0] + oe * p.P0_l * (__expf(p.lam_l * v[0]) - 1.0f);
    float P_rv = e * p.E_rv * v[3] + oe * p.P0_r * (__expf(p.lam_r * v[3]) - 1.0f);
    float P_ao = p.E_ao * v[1];
    float P_vc = p.E_vc * v[2];
    float P_pa = p.E_pa * v[4];
    float P_pu = p.E_pu * v[5];
    float Q_mt  = fmaxf((P_pu - P_lv) * p.rR_mt, 0.0f);
    float Q_av  = fmaxf((P_lv - P_ao) * p.rR_av, 0.0f);
    float Q_sys =       (P_ao - P_vc) * p.rR_sys;
    float Q_tc  = fmaxf((P_vc - P_rv) * p.rR_tc, 0.0f);
    float Q_pv  = fmaxf((P_rv - P_pa) * p.rR_pv, 0.0f);
    float Q_pul =       (P_pa - P_pu) * p.rR_pul;
    d[0] = Q_mt  - Q_av;
    d[1] = Q_av  - Q_sys;
    d[2] = Q_sys - Q_tc;
    d[3] = Q_tc  - Q_pv;
    d[4] = Q_pv  - Q_pul;
    d[5] = Q_pul - Q_mt;
}

__global__ __launch_bounds__(32) void smith_ode_kernel(
    const float* __restrict__ y0, const float* __restrict__ prm,
    float* __restrict__ out, int B, int nIter)
{
    // double-buffered snapshot staging: [phase][lane*6+j] == global order (b*6+j)
    __shared__ float smem[2][32 * 6];

    const int lane = threadIdx.x;                 // 0..31 (wave32)
    const int b    = blockIdx.x * 32 + lane;      // system id

    SmithP p;
    {
        const float* q = prm + (size_t)b * 16;
        p.E_lv = q[0];  p.E_rv = q[1];  p.E_ao = q[2];
        p.E_vc = q[3];  p.E_pa = q[4];  p.E_pu = q[5];
        p.rR_mt  = 1.0f / q[6];   p.rR_av  = 1.0f / q[7];
        p.rR_sys = 1.0f / q[8];   p.rR_tc  = 1.0f / q[9];
        p.rR_pv  = 1.0f / q[10];  p.rR_pul = 1.0f / q[11];
        p.P0_l = q[12]; p.lam_l = q[13]; p.P0_r = q[14]; p.lam_r = q[15];
    }

    float v[6];
#pragma unroll
    for (int j = 0; j < 6; ++j) v[j] = y0[(size_t)b * 6 + j];

    const uint64_t outBase   = (uint64_t)(uintptr_t)out;
    const uint64_t rowBytes  = (uint64_t)B * 6u * 4u;          // bytes per snapshot row
    const uint64_t blkOffset = (uint64_t)blockIdx.x * 192u * 4u + (uint64_t)(lane * 8);

    // Stage this block's 32x6 snapshot in LDS, then async-store 3x256B packets.
    auto snapshot = [&](int s) {
        float* buf = smem[s & 1];
        // buffer (s-2) used this LDS region; its 3 async ops are the oldest in
        // flight. Allow the 3 ops of snapshot (s-1) to remain outstanding.
        asm volatile("s_wait_asynccnt 0x3" ::: "memory");
#pragma unroll
        for (int j = 0; j < 6; ++j) buf[lane * 6 + j] = v[j];   // ds_store, conflict-free
        // async engine reads LDS: our DS writes must have landed first
        asm volatile("s_wait_dscnt 0x0" ::: "memory");
        // low 32 bits of a flat LDS pointer are the LDS byte offset (aperture in [63:32])
        uint32_t lbase = (uint32_t)(uintptr_t)buf + (uint32_t)(lane * 8);
        uint64_t gbase = outBase + (uint64_t)s * rowBytes + blkOffset;
#pragma unroll
        for (int c = 0; c < 3; ++c) {
            uint64_t ga = gbase + (uint64_t)(c * 256);
            uint32_t la = lbase + (uint32_t)(c * 256);
            // CDNA5 async LDS->global store, 8B/lane = 256B/wave, ASYNCcnt-tracked
            asm volatile("global_store_async_from_lds_b64 %0, %1, off"
                         :: "v"(ga), "v"(la) : "memory");
        }
    };

    snapshot(0);                                  // ys[0] == y0

    float t = 0.0f;
    for (int i = 0; i < nIter; ++i) {
        float e0 = smith_driver(t);               // reused across substeps
#pragma unroll 4
        for (int k = 0; k < ODE_SUBSTEPS; ++k) {
            float tk = t + (float)k * ODE_DT;
            float e1 = smith_driver(tk + ODE_DT); // driver at end of substep
            float k1[6], yt[6], k2[6];
            smith_vf(e0, v, k1, p);
#pragma unroll
            for (int j = 0; j < 6; ++j) yt[j] = v[j] + ODE_DT * k1[j];
            smith_vf(e1, yt, k2, p);
#pragma unroll
            for (int j = 0; j < 6; ++j) v[j] = v[j] + (0.5f * ODE_DT) * (k1[j] + k2[j]);
            e0 = e1;
        }
        t += 0.04f;                               // matches float32 accumulation in reference
        snapshot(i + 1);
    }
    // s_endpgm performs an implicit wait-idle, draining ASYNCcnt.
}

extern "C" void kernel_launch(void* const* d_in, const int* in_sizes, int n_in,
                              void* d_out, int out_size, void* d_ws, size_t ws_size,
                              hipStream_t stream)
{
    const float* y0  = (const float*)d_in[0];   // [B,6]
    const float* prm = (const float*)d_in[1];   // [B,16]
    // d_in[2] = ts; only its length matters (save count)
    float* out = (float*)d_out;                 // [n_ts, B, 6]

    int B     = in_sizes[0] / 6;                // 2048
    int nIter = in_sizes[2] - 1;                // 1000

    dim3 grid(B / 32), block(32);               // 64 single-wave workgroups
    smith_ode_kernel<<<grid, block, 0, stream>>>(y0, prm, out, B, nIter);
}